// UnaryEncoder_22445499089473
// MI455X (gfx1250) — compile-verified
//
#include <hip/hip_runtime.h>
#include <hip/hip_bf16.h>

#define IN_DIM   128
#define HIDDEN   128
#define HEADS    8
#define CH1      16      // HIDDEN / HEADS
#define NCLS     64
#define NEG_SLOPE 0.2f
#define LN_EPS   1e-5f

typedef float v2f __attribute__((ext_vector_type(2)));
typedef float v8f __attribute__((ext_vector_type(8)));

// ---- monotone float<->uint encoding for atomicMax-based segment max ----
__device__ __forceinline__ unsigned enc_f32(float f) {
    unsigned u = __float_as_uint(f);
    return (u & 0x80000000u) ? ~u : (u | 0x80000000u);
}
__device__ __forceinline__ float dec_f32(unsigned u) {
    return (u & 0x80000000u) ? __uint_as_float(u & 0x7fffffffu)
                             : __uint_as_float(~u);
}
#define ENC_NEGINF 0x007FFFFFu   // enc(-inf)

__device__ __forceinline__ void edge_sd(int e, int E, const int* __restrict__ ei,
                                        int& s, int& d) {
    if (e < E) { s = ei[e]; d = ei[E + e]; }
    else       { s = e - E; d = s; }          // appended self-loops
}

// ---------------- init: accumulators + logits seeded with bias2 -------------
__global__ void k_init(unsigned* __restrict__ m1, float* __restrict__ s1,
                       unsigned* __restrict__ m2, float* __restrict__ s2,
                       float* __restrict__ logits, const float* __restrict__ bias2,
                       int n) {
    int i = blockIdx.x * blockDim.x + threadIdx.x;
    if (i < n * HEADS) { m1[i] = ENC_NEGINF; s1[i] = 0.f; }
    if (i < n)         { m2[i] = ENC_NEGINF; s2[i] = 0.f; }
    if (i < n * NCLS)  logits[i] = bias2[i & (NCLS - 1)];
}

// ---- GEMM1: x @ [W1 | res_W] ; waves 0-3 -> h1, waves 4-7 -> agg1(+bias1) --
__global__ __launch_bounds__(256) void k_gemm1(
    const float* __restrict__ x, const float* __restrict__ W1,
    const float* __restrict__ resW, const float* __restrict__ bias1,
    float* __restrict__ h1, float* __restrict__ agg1, int n)
{
    const int wave  = threadIdx.x >> 5;
    const int lane  = threadIdx.x & 31;
    const int row0  = blockIdx.x * 16;
    const int m     = lane & 15;
    const int khalf = (lane >> 4) * 2;
    const int ct0   = wave * 32;                 // global col base 0..255
    const bool isRes = (ct0 >= HIDDEN);
    const float* W  = isRes ? resW : W1;
    const int c0    = isRes ? (ct0 - HIDDEN) : ct0;

    int arow_idx = row0 + m; if (arow_idx >= n) arow_idx = n - 1;
    const float* arow = x + (size_t)arow_idx * IN_DIM + khalf;

    v8f acc0 = {}; v8f acc1 = {};
    for (int kk = 0; kk < IN_DIM; kk += 4) {
        v2f a = *(const v2f*)(arow + kk);
        const float* br = W + (size_t)(kk + khalf) * HIDDEN;
        v2f b0 = { br[c0 + m],       br[HIDDEN + c0 + m] };
        v2f b1 = { br[c0 + 16 + m],  br[HIDDEN + c0 + 16 + m] };
        acc0 = __builtin_amdgcn_wmma_f32_16x16x4_f32(false, a, false, b0,
                                                     (short)0, acc0, false, false);
        acc1 = __builtin_amdgcn_wmma_f32_16x16x4_f32(false, a, false, b1,
                                                     (short)0, acc1, false, false);
    }

    const int nloc = lane & 15;
    const int mhi  = (lane >> 4) * 8;
    #pragma unroll
    for (int j = 0; j < 8; ++j) {
        int row = row0 + mhi + j;
        if (row >= n) continue;
        if (!isRes) {
            h1[(size_t)row * HIDDEN + ct0 + nloc]      = acc0[j];
            h1[(size_t)row * HIDDEN + ct0 + 16 + nloc] = acc1[j];
        } else {
            int cA = c0 + nloc, cB = c0 + 16 + nloc;
            agg1[(size_t)row * HIDDEN + cA] = acc0[j] + bias1[cA];
            agg1[(size_t)row * HIDDEN + cB] = acc1[j] + bias1[cB];
        }
    }
}

// ---------------- per-node attention coefficients, layer 1 ------------------
__global__ void k_att1(const float* __restrict__ h1,
                       const float* __restrict__ att_s, const float* __restrict__ att_d,
                       float* __restrict__ as, float* __restrict__ ad, int n)
{
    int i = blockIdx.x * blockDim.x + threadIdx.x;   // n*HEADS + h
    if (i >= n * HEADS) return;
    int node = i >> 3, h = i & 7;
    const float* base = h1 + (size_t)node * HIDDEN + h * CH1;
    const float* vs = att_s + h * CH1;
    const float* vd = att_d + h * CH1;
    float s = 0.f, d = 0.f;
    #pragma unroll
    for (int c = 0; c < CH1; ++c) { float v = base[c]; s += v * vs[c]; d += v * vd[c]; }
    as[i] = s; ad[i] = d;
}

// ---------------- edge pass 1: segment max (layer 1) ------------------------
__global__ void k_max1(const int* __restrict__ ei, const float* __restrict__ as,
                       const float* __restrict__ ad, unsigned* __restrict__ m1,
                       int E, int tot)
{
    int e = blockIdx.x * blockDim.x + threadIdx.x;
    if (e >= tot) return;
    int s, d; edge_sd(e, E, ei, s, d);
    const float4* ps = (const float4*)(as + (size_t)s * 8);
    const float4* pd = (const float4*)(ad + (size_t)d * 8);
    float4 s0 = ps[0], s1v = ps[1], d0 = pd[0], d1 = pd[1];
    float t[8] = { s0.x + d0.x, s0.y + d0.y, s0.z + d0.z, s0.w + d0.w,
                   s1v.x + d1.x, s1v.y + d1.y, s1v.z + d1.z, s1v.w + d1.w };
    #pragma unroll
    for (int h = 0; h < 8; ++h) {
        float v = t[h]; v = v > 0.f ? v : NEG_SLOPE * v;
        atomicMax(m1 + (size_t)d * 8 + h, enc_f32(v));
    }
}

// ---------------- edge pass 2: exp + segment sum (layer 1) ------------------
__global__ void k_sum1(const int* __restrict__ ei, const float* __restrict__ as,
                       const float* __restrict__ ad, const unsigned* __restrict__ m1,
                       float* __restrict__ s1, int E, int tot)
{
    int e = blockIdx.x * blockDim.x + threadIdx.x;
    if (e >= tot) return;
    int s, d; edge_sd(e, E, ei, s, d);
    const float4* ps = (const float4*)(as + (size_t)s * 8);
    const float4* pd = (const float4*)(ad + (size_t)d * 8);
    float4 s0 = ps[0], s1v = ps[1], d0 = pd[0], d1 = pd[1];
    float t[8] = { s0.x + d0.x, s0.y + d0.y, s0.z + d0.z, s0.w + d0.w,
                   s1v.x + d1.x, s1v.y + d1.y, s1v.z + d1.z, s1v.w + d1.w };
    #pragma unroll
    for (int h = 0; h < 8; ++h) {
        float v = t[h]; v = v > 0.f ? v : NEG_SLOPE * v;
        float w = expf(v - dec_f32(m1[(size_t)d * 8 + h]));
        atomicAdd(&s1[(size_t)d * 8 + h], w);
    }
}

// ----- edge pass 3: weighted message scatter (layer 1), one wave per edge ---
__global__ __launch_bounds__(256) void k_msg1(
    const int* __restrict__ ei, const float* __restrict__ as,
    const float* __restrict__ ad, const unsigned* __restrict__ m1,
    const float* __restrict__ s1, const float* __restrict__ h1,
    float* __restrict__ agg1, int E, int tot)
{
    int wid = blockIdx.x * (blockDim.x >> 5) + (threadIdx.x >> 5);
    if (wid >= tot) return;
    int lane = threadIdx.x & 31;
    int s, d; edge_sd(wid, E, ei, s, d);
    int head = lane >> 2;                       // lane covers features 4l..4l+3
    float t = as[(size_t)s * 8 + head] + ad[(size_t)d * 8 + head];
    t = t > 0.f ? t : NEG_SLOPE * t;
    float coeff = expf(t - dec_f32(m1[(size_t)d * 8 + head]))
                  / (s1[(size_t)d * 8 + head] + 1e-16f);
    float4 hv = *(const float4*)(h1 + (size_t)s * HIDDEN + lane * 4);
    float* o = agg1 + (size_t)d * HIDDEN + lane * 4;
    atomicAdd(o + 0, hv.x * coeff);
    atomicAdd(o + 1, hv.y * coeff);
    atomicAdd(o + 2, hv.z * coeff);
    atomicAdd(o + 3, hv.w * coeff);
}

// ---------------- LayerNorm + ELU, one wave per node ------------------------
__global__ __launch_bounds__(256) void k_ln(const float* __restrict__ agg1,
                                            const float* __restrict__ gamma,
                                            const float* __restrict__ beta,
                                            float* __restrict__ hout, int n)
{
    int node = blockIdx.x * (blockDim.x >> 5) + (threadIdx.x >> 5);
    if (node >= n) return;
    int lane = threadIdx.x & 31;
    float4 v = *(const float4*)(agg1 + (size_t)node * HIDDEN + lane * 4);
    float sum = v.x + v.y + v.z + v.w;
    for (int off = 16; off >= 1; off >>= 1) sum += __shfl_xor(sum, off, 32);
    float mu = sum * (1.f / HIDDEN);
    float4 dv = { v.x - mu, v.y - mu, v.z - mu, v.w - mu };
    float ss = dv.x * dv.x + dv.y * dv.y + dv.z * dv.z + dv.w * dv.w;
    for (int off = 16; off >= 1; off >>= 1) ss += __shfl_xor(ss, off, 32);
    float rstd = rsqrtf(ss * (1.f / HIDDEN) + LN_EPS);
    float4 g = *(const float4*)(gamma + lane * 4);
    float4 b = *(const float4*)(beta + lane * 4);
    float4 y = { dv.x * rstd * g.x + b.x, dv.y * rstd * g.y + b.y,
                 dv.z * rstd * g.z + b.z, dv.w * rstd * g.w + b.w };
    y.x = y.x > 0.f ? y.x : expm1f(y.x);
    y.y = y.y > 0.f ? y.y : expm1f(y.y);
    y.z = y.z > 0.f ? y.z : expm1f(y.z);
    y.w = y.w > 0.f ? y.w : expm1f(y.w);
    *(float4*)(hout + (size_t)node * HIDDEN + lane * 4) = y;
}

// ---------------- GEMM2: h @ W2 (128 -> 64), wave per 16x16 tile ------------
__global__ __launch_bounds__(128) void k_gemm2(const float* __restrict__ h,
                                               const float* __restrict__ W2,
                                               float* __restrict__ h2, int n)
{
    const int wave  = threadIdx.x >> 5;
    const int lane  = threadIdx.x & 31;
    const int row0  = blockIdx.x * 16;
    const int m     = lane & 15;
    const int khalf = (lane >> 4) * 2;
    const int c0    = wave * 16;

    int arow_idx = row0 + m; if (arow_idx >= n) arow_idx = n - 1;
    const float* arow = h + (size_t)arow_idx * HIDDEN + khalf;

    v8f acc = {};
    for (int kk = 0; kk < HIDDEN; kk += 4) {
        v2f a = *(const v2f*)(arow + kk);
        const float* br = W2 + (size_t)(kk + khalf) * NCLS;
        v2f b = { br[c0 + m], br[NCLS + c0 + m] };
        acc = __builtin_amdgcn_wmma_f32_16x16x4_f32(false, a, false, b,
                                                    (short)0, acc, false, false);
    }
    const int mhi = (lane >> 4) * 8;
    #pragma unroll
    for (int j = 0; j < 8; ++j) {
        int row = row0 + mhi + j;
        if (row < n) h2[(size_t)row * NCLS + c0 + m] = acc[j];
    }
}

// ---------------- per-node attention coefficients, layer 2 ------------------
__global__ void k_att2(const float* __restrict__ h2, const float* __restrict__ att_s,
                       const float* __restrict__ att_d, float* __restrict__ as,
                       float* __restrict__ ad, int n)
{
    int node = blockIdx.x * blockDim.x + threadIdx.x;
    if (node >= n) return;
    const float* base = h2 + (size_t)node * NCLS;
    float s = 0.f, d = 0.f;
    #pragma unroll 8
    for (int c = 0; c < NCLS; ++c) { float v = base[c]; s += v * att_s[c]; d += v * att_d[c]; }
    as[node] = s; ad[node] = d;
}

__global__ void k_max2(const int* __restrict__ ei, const float* __restrict__ as,
                       const float* __restrict__ ad, unsigned* __restrict__ m2,
                       int E, int tot)
{
    int e = blockIdx.x * blockDim.x + threadIdx.x;
    if (e >= tot) return;
    int s, d; edge_sd(e, E, ei, s, d);
    float t = as[s] + ad[d]; t = t > 0.f ? t : NEG_SLOPE * t;
    atomicMax(m2 + d, enc_f32(t));
}

__global__ void k_sum2(const int* __restrict__ ei, const float* __restrict__ as,
                       const float* __restrict__ ad, const unsigned* __restrict__ m2,
                       float* __restrict__ s2, int E, int tot)
{
    int e = blockIdx.x * blockDim.x + threadIdx.x;
    if (e >= tot) return;
    int s, d; edge_sd(e, E, ei, s, d);
    float t = as[s] + ad[d]; t = t > 0.f ? t : NEG_SLOPE * t;
    atomicAdd(&s2[d], expf(t - dec_f32(m2[d])));
}

__global__ __launch_bounds__(256) void k_msg2(
    const int* __restrict__ ei, const float* __restrict__ as,
    const float* __restrict__ ad, const unsigned* __restrict__ m2,
    const float* __restrict__ s2, const float* __restrict__ h2,
    float* __restrict__ logits, int E, int tot)
{
    int wid = blockIdx.x * (blockDim.x >> 5) + (threadIdx.x >> 5);
    if (wid >= tot) return;
    int lane = threadIdx.x & 31;
    int s, d; edge_sd(wid, E, ei, s, d);
    float t = as[s] + ad[d]; t = t > 0.f ? t : NEG_SLOPE * t;
    float coeff = expf(t - dec_f32(m2[d])) / (s2[d] + 1e-16f);
    float2 hv = *(const float2*)(h2 + (size_t)s * NCLS + lane * 2);
    float* o = logits + (size_t)d * NCLS + lane * 2;
    atomicAdd(o + 0, hv.x * coeff);
    atomicAdd(o + 1, hv.y * coeff);
}

extern "C" void kernel_launch(void* const* d_in, const int* in_sizes, int n_in,
                              void* d_out, int out_size, void* d_ws, size_t ws_size,
                              hipStream_t stream)
{
    const float* x        = (const float*)d_in[0];
    const int*   ei       = (const int*)  d_in[1];
    const float* W1       = (const float*)d_in[2];
    const float* att_src1 = (const float*)d_in[3];
    const float* att_dst1 = (const float*)d_in[4];
    const float* bias1    = (const float*)d_in[5];
    const float* resW     = (const float*)d_in[6];
    const float* gamma    = (const float*)d_in[7];
    const float* beta     = (const float*)d_in[8];
    const float* W2       = (const float*)d_in[9];
    const float* att_src2 = (const float*)d_in[10];
    const float* att_dst2 = (const float*)d_in[11];
    const float* bias2    = (const float*)d_in[12];

    const int n   = in_sizes[0] / IN_DIM;   // 100000
    const int E   = in_sizes[1] / 2;        // 1600000
    const int tot = E + n;                  // + self-loops

    float* hout   = (float*)d_out;                       // [n,128] final h (also temp h1)
    float* logits = hout + (size_t)n * HIDDEN;           // [n,64]

    float*    agg1 = (float*)d_ws;                       // [n,128]; reused as h2 [n,64]
    float*    as1  = agg1 + (size_t)n * HIDDEN;          // [n,8]
    float*    ad1  = as1  + (size_t)n * HEADS;           // [n,8]
    unsigned* m1   = (unsigned*)(ad1 + (size_t)n * HEADS);
    float*    s1   = (float*)(m1 + (size_t)n * HEADS);
    float*    as2  = s1 + (size_t)n * HEADS;             // [n]
    float*    ad2  = as2 + n;
    unsigned* m2   = (unsigned*)(ad2 + n);
    float*    s2   = (float*)(m2 + n);
    float*    h1   = hout;    // stage h1 in d_out; overwritten later by LN output
    float*    h2   = agg1;    // reuse agg1 after LayerNorm consumed it

    const int B = 256;
    k_init <<<(n * NCLS + B - 1) / B, B, 0, stream>>>(m1, s1, m2, s2, logits, bias2, n);
    k_gemm1<<<(n + 15) / 16, 256, 0, stream>>>(x, W1, resW, bias1, h1, agg1, n);
    k_att1 <<<(n * HEADS + B - 1) / B, B, 0, stream>>>(h1, att_src1, att_dst1, as1, ad1, n);
    k_max1 <<<(tot + B - 1) / B, B, 0, stream>>>(ei, as1, ad1, m1, E, tot);
    k_sum1 <<<(tot + B - 1) / B, B, 0, stream>>>(ei, as1, ad1, m1, s1, E, tot);
    k_msg1 <<<(tot + 7) / 8, 256, 0, stream>>>(ei, as1, ad1, m1, s1, h1, agg1, E, tot);
    k_ln   <<<(n + 7) / 8, 256, 0, stream>>>(agg1, gamma, beta, hout, n);
    k_gemm2<<<(n + 15) / 16, 128, 0, stream>>>(hout, W2, h2, n);
    k_att2 <<<(n + B - 1) / B, B, 0, stream>>>(h2, att_src2, att_dst2, as2, ad2, n);
    k_max2 <<<(tot + B - 1) / B, B, 0, stream>>>(ei, as2, ad2, m2, E, tot);
    k_sum2 <<<(tot + B - 1) / B, B, 0, stream>>>(ei, as2, ad2, m2, s2, E, tot);
    k_msg2 <<<(tot + 7) / 8, 256, 0, stream>>>(ei, as2, ad2, m2, s2, h2, logits, E, tot);

    (void)n_in; (void)out_size; (void)ws_size; (void)att_dst2;
}